// Attention_52243982188856
// MI455X (gfx1250) — compile-verified
//
#include <hip/hip_runtime.h>
#include <math.h>

typedef __attribute__((ext_vector_type(2))) float v2f;
typedef __attribute__((ext_vector_type(4))) float v4f;
typedef __attribute__((ext_vector_type(8))) float v8f;

#define HEAD_DIM 128
#define N_KV 8
#define N_REP 8
#define N_HEADS 64
#define BSZ 32
#define DMODEL 8192
#define NQKV 10240
#define SPLITK 4
#define ASPLIT 4

// ---------------------------------------------------------------------------
// GEMM: C_partial[split][32][N] = A[32][Kslice] x B   (fp32, WMMA 16x16x4)
// TRANSB=0: B[k][n] = W[k][n] (KxN row-major);  TRANSB=1: B[k][n] = W[n][k].
// 256 threads = 8 waves; C tile 32x64; K-tile 32 (8 WMMAs / tile).
// Double-buffered LDS + register staging: next tile's global loads are in
// flight while the current tile's WMMAs run. Weights use non-temporal loads.
// ---------------------------------------------------------------------------
template <int TRANSB>
__global__ __launch_bounds__(256)
void gemm32_wmma(const float* __restrict__ A, const float* __restrict__ W,
                 float* __restrict__ Cp, int N, int K)
{
  __shared__ __align__(16) float As[2][32][36];   // 36: 16B-aligned rows, no conflicts
  __shared__ __align__(16) float Bs[2][32][68];
  const int n0    = blockIdx.x * 64;
  const int split = blockIdx.y;
  const int kSlice = K / SPLITK;
  const int kBeg   = split * kSlice;
  const int ntiles = kSlice / 32;
  const int tid  = threadIdx.x;
  const int wave = tid >> 5;
  const int lane = tid & 31;
  const int half = lane >> 4;          // 0: lanes 0-15 (K=k,k+1), 1: lanes 16-31 (K=k+2,k+3)
  const int l    = lane & 15;
  const int wm   = (wave >> 2) << 4;   // 0 / 16
  const int wn   = (wave & 3) << 4;    // 0..48

  // per-thread load coordinates (no predicates: exact coverage)
  const int am  = tid >> 3;            // A: 32 rows x 32 k  -> 256 float4
  const int ak4 = (tid & 7) << 2;
  v4f ra, rb0, rb1;

  auto loadTile = [&](int k0) {
    ra = *(const v4f*)(A + (size_t)am * K + k0 + ak4);
    if (!TRANSB) {
      const int kk = tid >> 4, nn = (tid & 15) << 2;
      rb0 = __builtin_nontemporal_load((const v4f*)(W + (size_t)(k0 + kk) * N + n0 + nn));
      rb1 = __builtin_nontemporal_load((const v4f*)(W + (size_t)(k0 + kk + 16) * N + n0 + nn));
    } else {
      const int nn = tid >> 3, kk = (tid & 7) << 2;
      rb0 = __builtin_nontemporal_load((const v4f*)(W + (size_t)(n0 + nn) * K + k0 + kk));
      rb1 = __builtin_nontemporal_load((const v4f*)(W + (size_t)(n0 + nn + 32) * K + k0 + kk));
    }
  };
  auto storeTile = [&](int buf) {
    *(v4f*)&As[buf][am][ak4] = ra;
    if (!TRANSB) {
      const int kk = tid >> 4, nn = (tid & 15) << 2;
      *(v4f*)&Bs[buf][kk][nn]      = rb0;
      *(v4f*)&Bs[buf][kk + 16][nn] = rb1;
    } else {
      const int nn = tid >> 3, kk = (tid & 7) << 2;
      Bs[buf][kk + 0][nn] = rb0[0]; Bs[buf][kk + 1][nn] = rb0[1];
      Bs[buf][kk + 2][nn] = rb0[2]; Bs[buf][kk + 3][nn] = rb0[3];
      Bs[buf][kk + 0][nn + 32] = rb1[0]; Bs[buf][kk + 1][nn + 32] = rb1[1];
      Bs[buf][kk + 2][nn + 32] = rb1[2]; Bs[buf][kk + 3][nn + 32] = rb1[3];
    }
  };

  v8f c = {0.f, 0.f, 0.f, 0.f, 0.f, 0.f, 0.f, 0.f};

  loadTile(kBeg);
  storeTile(0);
  __syncthreads();

  for (int t = 0; t < ntiles; ++t) {
    const int buf = t & 1;
    if (t + 1 < ntiles) loadTile(kBeg + (t + 1) * 32);  // in flight during compute
#pragma unroll
    for (int kk = 0; kk < 32; kk += 4) {
      v2f a, b;
      a[0] = As[buf][wm + l][kk + half * 2 + 0];
      a[1] = As[buf][wm + l][kk + half * 2 + 1];
      b[0] = Bs[buf][kk + half * 2 + 0][wn + l];
      b[1] = Bs[buf][kk + half * 2 + 1][wn + l];
      c = __builtin_amdgcn_wmma_f32_16x16x4_f32(false, a, false, b,
                                                (short)0, c, false, false);
    }
    if (t + 1 < ntiles) storeTile(buf ^ 1);             // loadcnt wait lands here
    __syncthreads();
  }

  float* Cout = Cp + (size_t)split * 32 * N;
#pragma unroll
  for (int i = 0; i < 8; ++i) {
    const int m = wm + (half << 3) + i;   // VGPR i -> row i (lanes 0-15) / i+8
    Cout[(size_t)m * N + n0 + wn + l] = c[i];
  }
}

// Deterministic split-K reduction (no fp atomics -> bit-identical replays)
__global__ __launch_bounds__(256)
void reduce_splitk(const float* __restrict__ p, float* __restrict__ o, int total)
{
  int i = blockIdx.x * 256 + threadIdx.x;
  if (i < total)
    o[i] = (p[i] + p[(size_t)total + i]) +
           (p[(size_t)2 * total + i] + p[(size_t)3 * total + i]);
}

// ---------------------------------------------------------------------------
// RoPE on q and new-k, plus v copy-out (input caches never mutated).
// qkv row layout (per batch, 8 groups of 1280): [q(8x128) | k(128) | v(128)]
// ---------------------------------------------------------------------------
__global__ __launch_bounds__(128)
void rope_split(const float* __restrict__ qkv, const float* __restrict__ fcos,
                const float* __restrict__ fsin, float* __restrict__ qout,
                float* __restrict__ kout, float* __restrict__ vout)
{
  const int row = blockIdx.x;       // b*80 + j
  const int b = row / 80;
  const int j = row % 80;
  const int d = threadIdx.x;        // 0..127
  const float* base = qkv + (size_t)b * NQKV;
  if (j < 64) {                     // q head j
    const int g = j >> 3, r = j & 7;
    const float* src = base + g * 1280 + r * 128;
    const int p = d >> 1;
    const float t0 = src[p * 2], t1 = src[p * 2 + 1];
    const float cc = fcos[p], ss = fsin[p];
    qout[((size_t)b * 64 + j) * 128 + d] =
        (d & 1) ? (t0 * ss + t1 * cc) : (t0 * cc - t1 * ss);
  } else if (j < 72) {              // k, kv-head j-64
    const int g = j - 64;
    const float* src = base + g * 1280 + 1024;
    const int p = d >> 1;
    const float t0 = src[p * 2], t1 = src[p * 2 + 1];
    const float cc = fcos[p], ss = fsin[p];
    kout[((size_t)b * 8 + g) * 128 + d] =
        (d & 1) ? (t0 * ss + t1 * cc) : (t0 * cc - t1 * ss);
  } else {                          // v, kv-head j-72
    const int g = j - 72;
    vout[((size_t)b * 8 + g) * 128 + d] = base[g * 1280 + 1152 + d];
  }
}

// ---------------------------------------------------------------------------
// Flash-decode attention, T split across ASPLIT workgroups (grid 32x8x4 =
// 1024 WGs to saturate HBM). Wave = rep row; lane = key within 32-key tile.
// Register-staged KV prefetch: next tile's non-temporal global loads are
// issued before the current tile's score/accumulate phase.
// Row t == start_pos comes from the freshly computed k/v.
// Emits unnormalized partials (acc, running max m, running sum s).
// ---------------------------------------------------------------------------
__global__ __launch_bounds__(256)
void attn_decode(const float* __restrict__ q, const float* __restrict__ ck,
                 const float* __restrict__ cv, const float* __restrict__ kn,
                 const float* __restrict__ vn, const float* __restrict__ mask,
                 float* __restrict__ pacc, float* __restrict__ pm,
                 float* __restrict__ ps, int T, int start_pos)
{
  __shared__ __align__(16) float Qs[8][128];
  __shared__ __align__(16) float Ks[32][132];  // 132: 16B-aligned rows
  __shared__ __align__(16) float Vs[32][132];
  const int b = blockIdx.x, g = blockIdx.y, sp = blockIdx.z;
  const int tid = threadIdx.x, wave = tid >> 5, lane = tid & 31;
  const int Tslice = T / ASPLIT;
  const int tBeg   = sp * Tslice;
  const int ntiles = Tslice / 32;

  for (int i = tid; i < 8 * 128; i += 256) {
    const int r = i >> 7, d = i & 127;
    Qs[r][d] = q[((size_t)b * 64 + g * 8 + r) * 128 + d];
  }

  const int lr = tid >> 5;         // staged row base (0..7), + pass*8
  const int ld = (tid & 31) << 2;  // dim offset (float4)
  v4f rk[4], rv[4];

  auto loadTile = [&](int t0) {
#pragma unroll
    for (int pass = 0; pass < 4; ++pass) {
      const int t = t0 + lr + pass * 8;
      const float *ksp, *vsp;
      if (t == start_pos) {
        ksp = kn + (((size_t)b * 8 + g) * 128 + ld);
        vsp = vn + (((size_t)b * 8 + g) * 128 + ld);
      } else {
        ksp = ck + ((((size_t)b * T + t) * 8 + g) * 128 + ld);
        vsp = cv + ((((size_t)b * T + t) * 8 + g) * 128 + ld);
      }
      rk[pass] = __builtin_nontemporal_load((const v4f*)ksp);
      rv[pass] = __builtin_nontemporal_load((const v4f*)vsp);
    }
  };

  loadTile(tBeg);
  float m_run = -1e30f, s_run = 0.f;
  v4f acc = {0.f, 0.f, 0.f, 0.f};
  const float scale = 0.08838834764831845f;  // 1/sqrt(128)

  for (int tt = 0; tt < ntiles; ++tt) {
    __syncthreads();                    // prior reads of Ks/Vs (and Qs init) done
#pragma unroll
    for (int pass = 0; pass < 4; ++pass) {
      const int r = lr + pass * 8;
      *(v4f*)&Ks[r][ld] = rk[pass];
      *(v4f*)&Vs[r][ld] = rv[pass];
    }
    __syncthreads();
    const int t0 = tBeg + tt * 32;
    if (tt + 1 < ntiles) loadTile(t0 + 32);   // prefetch overlaps compute

    // score: this wave's q row vs. 32 keys (one per lane), b128 LDS reads
    float s = 0.f;
#pragma unroll
    for (int i = 0; i < 128; i += 4) {
      const v4f qv = *(const v4f*)&Qs[wave][i];
      const v4f kk = *(const v4f*)&Ks[lane][i];
      s += qv[0] * kk[0] + qv[1] * kk[1] + qv[2] * kk[2] + qv[3] * kk[3];
    }
    s = s * scale + mask[t0 + lane];

    float tmax = s;
    for (int off = 16; off; off >>= 1) tmax = fmaxf(tmax, __shfl_xor(tmax, off, 32));
    const float m_new = fmaxf(m_run, tmax);
    const float p = __expf(s - m_new);
    const float corr = __expf(m_run - m_new);
    float psum = p;
    for (int off = 16; off; off >>= 1) psum += __shfl_xor(psum, off, 32);
    s_run = s_run * corr + psum;
    acc[0] *= corr; acc[1] *= corr; acc[2] *= corr; acc[3] *= corr;

#pragma unroll
    for (int key = 0; key < 32; ++key) {
      const float pk = __shfl(p, key, 32);
      const v4f vv = *(const v4f*)&Vs[key][ld];
      acc[0] = fmaf(pk, vv[0], acc[0]);
      acc[1] = fmaf(pk, vv[1], acc[1]);
      acc[2] = fmaf(pk, vv[2], acc[2]);
      acc[3] = fmaf(pk, vv[3], acc[3]);
    }
    m_run = m_new;
  }

  const size_t pidx = (((size_t)sp * BSZ + b) * N_KV + g) * N_REP + wave;
  *(v4f*)&pacc[pidx * 128 + ld] = acc;
  if (lane == 0) { pm[pidx] = m_run; ps[pidx] = s_run; }
}

// Deterministic log-sum-exp combine of the ASPLIT partials.
__global__ __launch_bounds__(128)
void attn_combine(const float* __restrict__ pacc, const float* __restrict__ pm,
                  const float* __restrict__ ps, float* __restrict__ out)
{
  const int bh = blockIdx.x;          // b*64 + h
  const int b = bh >> 6, h = bh & 63;
  const int g = h >> 3, r = h & 7;
  const int d = threadIdx.x;
  float m[ASPLIT], sv[ASPLIT], M = -1e30f;
#pragma unroll
  for (int i = 0; i < ASPLIT; ++i) {
    const size_t pidx = (((size_t)i * BSZ + b) * N_KV + g) * N_REP + r;
    m[i] = pm[pidx]; sv[i] = ps[pidx];
    M = fmaxf(M, m[i]);
  }
  float denom = 0.f, a = 0.f;
#pragma unroll
  for (int i = 0; i < ASPLIT; ++i) {
    const float w = __expf(m[i] - M);
    const size_t pidx = (((size_t)i * BSZ + b) * N_KV + g) * N_REP + r;
    denom += sv[i] * w;
    a += w * pacc[pidx * 128 + d];
  }
  out[(size_t)bh * 128 + d] = a / denom;
}

// ---------------------------------------------------------------------------
extern "C" void kernel_launch(void* const* d_in, const int* in_sizes, int n_in,
                              void* d_out, int out_size, void* d_ws, size_t ws_size,
                              hipStream_t stream)
{
  (void)n_in; (void)out_size; (void)ws_size;
  const float* x       = (const float*)d_in[0];
  const float* cache_k = (const float*)d_in[1];
  const float* cache_v = (const float*)d_in[2];
  const float* wqkv    = (const float*)d_in[3];
  const float* wo      = (const float*)d_in[4];
  const float* fcos    = (const float*)d_in[5];
  const float* fsin    = (const float*)d_in[6];
  const float* mask    = (const float*)d_in[7];
  const int T = in_sizes[1] / (BSZ * N_KV * HEAD_DIM);   // cache_k = [32,T,8,128]
  const int start_pos = T - 1;                           // seqlen == 1

  float* ws       = (float*)d_ws;
  float* qkv_part = ws;                                           // 4*32*10240
  float* qkv      = qkv_part + (size_t)SPLITK * BSZ * NQKV;       // 32*10240
  float* qrope    = qkv + (size_t)BSZ * NQKV;                     // 32*64*128
  float* knew     = qrope + (size_t)BSZ * N_HEADS * HEAD_DIM;     // 32*8*128
  float* vnew     = knew + (size_t)BSZ * N_KV * HEAD_DIM;         // 32*8*128
  float* attn     = vnew + (size_t)BSZ * N_KV * HEAD_DIM;         // 32*8192
  float* out_part = attn + (size_t)BSZ * DMODEL;                  // 4*32*8192
  float* pacc     = out_part + (size_t)SPLITK * BSZ * DMODEL;     // 4*32*64*128
  float* pm       = pacc + (size_t)ASPLIT * BSZ * N_HEADS * HEAD_DIM; // 4*32*64
  float* ps       = pm + (size_t)ASPLIT * BSZ * N_HEADS;          // 4*32*64

  // 1) QKV projection: [32,10240] = x[32,8192] @ wqkv  (streams 335 MB once)
  gemm32_wmma<0><<<dim3(NQKV / 64, SPLITK), 256, 0, stream>>>(x, wqkv, qkv_part,
                                                              NQKV, DMODEL);
  reduce_splitk<<<(BSZ * NQKV + 255) / 256, 256, 0, stream>>>(qkv_part, qkv,
                                                              BSZ * NQKV);
  // 2) RoPE + split q/k/v
  rope_split<<<BSZ * 80, 128, 0, stream>>>(qkv, fcos, fsin, qrope, knew, vnew);

  // 3) Flash-decode attention, T-split x4 (1024 WGs stream 1.07 GB of KV)
  attn_decode<<<dim3(BSZ, N_KV, ASPLIT), 256, 0, stream>>>(
      qrope, cache_k, cache_v, knew, vnew, mask, pacc, pm, ps, T, start_pos);
  attn_combine<<<BSZ * N_HEADS, 128, 0, stream>>>(pacc, pm, ps, attn);

  // 4) Output projection: [32,8192] = attn @ wo^T  (streams 268 MB once)
  gemm32_wmma<1><<<dim3(DMODEL / 64, SPLITK), 256, 0, stream>>>(attn, wo, out_part,
                                                                DMODEL, DMODEL);
  reduce_splitk<<<(BSZ * DMODEL + 255) / 256, 256, 0, stream>>>(out_part,
                                                                (float*)d_out,
                                                                BSZ * DMODEL);
}